// GSP_DMPNN_71777493450840
// MI455X (gfx1250) — compile-verified
//
#include <hip/hip_runtime.h>
#include <hip/hip_bf16.h>
#include <math.h>

#define NN   25000
#define FF   128
#define EDIMM 16
#define NITER 4
#define EE   200000
#define LEE  800000
#define GGR  256
#define SNDD (6*FF)     /* 768 */
#define EPSV 1e-5f
#define ALPH 0.6f
#define NSLOPE 0.2f

typedef __attribute__((ext_vector_type(2))) float v2f;
typedef __attribute__((ext_vector_type(8))) float v8f;
typedef __attribute__((ext_vector_type(4))) int   v4i;

#if __has_builtin(__builtin_amdgcn_global_load_async_to_lds_b128)
#define HAVE_ASYNC_LDS 1
typedef __attribute__((address_space(1))) v4i as1_v4i;
typedef __attribute__((address_space(3))) v4i as3_v4i;
#endif

__device__ __forceinline__ float lrelu(float x){ return x >= 0.f ? x : NSLOPE * x; }

__device__ __forceinline__ void atomicMaxF(float* addr, float val){
  if (val >= 0.f) atomicMax((int*)addr, __float_as_int(val));
  else            atomicMin((unsigned int*)addr, __float_as_uint(val));
}

// ---------------------------------------------------------------------------
// WMMA f32 GEMM:  C[M,N] = A[M,K] @ W[K,N] (+bias) (opt tanh)
// Block tile 128(M) x 32(N), K staged in chunks of 32 through LDS.
// 8 waves / block; wave w computes rows [16w,16w+16) x cols [0,32).
// A tile staged via GLOBAL_LOAD_ASYNC_TO_LDS_B128 (ASYNCcnt) on full tiles;
// B tile staged synchronously but stored transposed so each lane's pair of
// B operands is contiguous (ds_load_2addr_b64, no v_mov shuffles).
// ---------------------------------------------------------------------------
#define BM 128
#define BN 32
#define KC 32

__global__ __launch_bounds__(256)
void k_gemm(const float* __restrict__ A, const float* __restrict__ W,
            const float* __restrict__ bias, float* __restrict__ C,
            int M, int K, int Nout, int op)
{
  __shared__ __align__(16) float As[BM][KC];   // [row][k]
  __shared__ __align__(16) float Bt[BN][KC];   // [col][k]  (transposed store)
  const int m0   = blockIdx.x * BM;
  const int n0   = blockIdx.y * BN;
  const int tid  = threadIdx.x;
  const int wave = tid >> 5;
  const int lane = tid & 31;
  const int lr   = lane & 15;
  const int hi   = lane >> 4;
  const int mbase = wave * 16;

  v8f acc0 = {}; v8f acc1 = {};

  for (int k0 = 0; k0 < K; k0 += KC) {
    const bool fullA = (m0 + BM <= M) && (k0 + KC <= K);   // uniform per block
#ifdef HAVE_ASYNC_LDS
    if (fullA) {
      // async DMA of the 128x32 A tile: 1024 float4's, 4 per thread
#pragma unroll
      for (int i = 0; i < 4; ++i) {
        int q = tid + i * 256;
        int r = q >> 3, c4 = q & 7;
        const float* gp = A + (size_t)(m0 + r) * K + k0 + c4 * 4;
        __builtin_amdgcn_global_load_async_to_lds_b128(
            (as1_v4i*)gp, (as3_v4i*)&As[r][c4 * 4], 0, 0);
      }
    } else
#endif
    {
#pragma unroll
      for (int i = 0; i < 16; ++i) {
        int idx = tid + i * 256;
        int r = idx >> 5, c = idx & 31;
        int gr = m0 + r, gc = k0 + c;
        As[r][c] = (gr < M && gc < K) ? A[(size_t)gr * K + gc] : 0.f;
      }
      if (k0 + KC < K) {           // prefetch next A K-tile (global_prefetch_b8)
        int pr = m0 + (tid & 127);
        if (pr < M) __builtin_prefetch(&A[(size_t)pr * K + k0 + KC], 0, 1);
      }
    }
    // B tile (sync, overlaps the in-flight async A copy), stored transposed
#pragma unroll
    for (int i = 0; i < 4; ++i) {
      int idx = tid + i * 256;
      int r = idx >> 5, c = idx & 31;          // r = k-offset, c = n-offset
      int gr = k0 + r, gc = n0 + c;
      Bt[c][r] = (gr < K && gc < Nout) ? W[(size_t)gr * Nout + gc] : 0.f;
    }
#ifdef HAVE_ASYNC_LDS
    if (fullA) {
#if __has_builtin(__builtin_amdgcn_s_wait_asynccnt)
      __builtin_amdgcn_s_wait_asynccnt(0);
#else
      asm volatile("s_wait_asynccnt 0" ::: "memory");
#endif
    }
#endif
    __syncthreads();
#pragma unroll
    for (int kk = 0; kk < KC; kk += 4) {
      const int ks = kk + 2 * hi;
      v2f a;  a.x  = As[mbase + lr][ks];  a.y  = As[mbase + lr][ks + 1];
      v2f b0; b0.x = Bt[lr][ks];          b0.y = Bt[lr][ks + 1];
      v2f b1; b1.x = Bt[16 + lr][ks];     b1.y = Bt[16 + lr][ks + 1];
#if __has_builtin(__builtin_amdgcn_wmma_f32_16x16x4_f32)
      acc0 = __builtin_amdgcn_wmma_f32_16x16x4_f32(false, a, false, b0, (short)0, acc0, false, false);
      acc1 = __builtin_amdgcn_wmma_f32_16x16x4_f32(false, a, false, b1, (short)0, acc1, false, false);
#else
      for (int r = 0; r < 8; ++r) { acc0[r] += a.x * b0.x + a.y * b0.y; acc1[r] += a.x * b1.x + a.y * b1.y; }
#endif
    }
    __syncthreads();
  }
#pragma unroll
  for (int r = 0; r < 8; ++r) {
    int row = m0 + mbase + r + 8 * hi;
    if (row < M) {
      int c0 = n0 + lr, c1 = n0 + 16 + lr;
      if (c0 < Nout) {
        float z = acc0[r] + (bias ? bias[c0] : 0.f);
        if (op == 1) z = tanhf(z);
        C[(size_t)row * Nout + c0] = z;
      }
      if (c1 < Nout) {
        float z = acc1[r] + (bias ? bias[c1] : 0.f);
        if (op == 1) z = tanhf(z);
        C[(size_t)row * Nout + c1] = z;
      }
    }
  }
}

// ---------------------------------------------------------------------------
// Elementwise / scatter helpers
// ---------------------------------------------------------------------------
__global__ void k_fill(float* p, float v, long long n){
  long long i = (long long)blockIdx.x * 256 + threadIdx.x;
  if (i < n) p[i] = v;
}
__global__ void k_copy(float* y, const float* x, long long n){
  long long i = (long long)blockIdx.x * 256 + threadIdx.x;
  if (i < n) y[i] = x[i];
}
__global__ void k_add(float* y, const float* a, const float* b, long long n){
  long long i = (long long)blockIdx.x * 256 + threadIdx.x;
  if (i < n) y[i] = a[i] + b[i];
}
__global__ void k_avg2(float* a, const float* b, long long n){  // a = (a+b)/2
  long long i = (long long)blockIdx.x * 256 + threadIdx.x;
  if (i < n) a[i] = (a[i] + b[i]) * 0.5f;
}
__global__ void k_build_ea(float* ea, const float* xu, const float* xv,
                           const int* ei0, const int* ei1){
  long long i = (long long)blockIdx.x * 256 + threadIdx.x;
  if (i < (long long)EE * FF) {
    int e = (int)(i / FF), f = (int)(i % FF);
    ea[i] = (ea[i] + xu[(size_t)ei0[e] * FF + f] + xv[(size_t)ei1[e] * FF + f]) * (1.f / 3.f);
  }
}
__global__ void k_deg(const int* dst, float* deg){
  int l = blockIdx.x * 256 + threadIdx.x;
  if (l < LEE) atomicAdd(&deg[dst[l]], 1.f);
}
__global__ void k_dinv(float* d){
  int e = blockIdx.x * 256 + threadIdx.x;
  if (e < EE) d[e] = rsqrtf(d[e] + 1.f);   // +1 self loop; always > 0
}
__global__ void k_scatter_feat(const int* src, const int* dst,
                               const float* x, float* y){
  long long i = (long long)blockIdx.x * 256 + threadIdx.x;
  if (i < (long long)LEE * FF) {
    int l = (int)(i / FF), f = (int)(i % FF);
    atomicAdd(&y[(size_t)dst[l] * FF + f], x[(size_t)src[l] * FF + f]);
  }
}

// warp(32)-per-row dual dot product
__global__ __launch_bounds__(256)
void k_rowdot2(const float* X, const float* w1, const float* w2,
               float* o1, float* o2, int M, int F){
  int row  = blockIdx.x * 8 + (threadIdx.x >> 5);
  int lane = threadIdx.x & 31;
  if (row >= M) return;
  float s1 = 0.f, s2 = 0.f;
  for (int f = lane; f < F; f += 32) {
    float v = X[(size_t)row * F + f];
    s1 += v * w1[f];
    if (w2) s2 += v * w2[f];
  }
  for (int off = 16; off; off >>= 1) {
    s1 += __shfl_down(s1, off);
    if (w2) s2 += __shfl_down(s2, off);
  }
  if (lane == 0) { o1[row] = s1; if (w2) o2[row] = s2; }
}

__global__ void k_gcn_scatter(const int* src, const int* dst,
                              const float* dinv, const float* hw, float* acc){
  int l = blockIdx.x * 256 + threadIdx.x;
  if (l < LEE) {
    int s = src[l], d = dst[l];
    atomicAdd(&acc[d], dinv[s] * dinv[d] * hw[s]);
  }
}
__global__ void k_score(float* sc, const float* gcnacc, const float* dinv,
                        const float* o1, const float* o2,
                        const float* gcn_b, const float* fbtl_b){
  int e = blockIdx.x * 256 + threadIdx.x;
  if (e < EE) {
    float g = gcnacc[e] + dinv[e] * dinv[e] * o1[e] + gcn_b[0];
    float f = o2[e] + fbtl_b[0];
    sc[e] = ALPH * g + (1.f - ALPH) * f;
  }
}
__global__ void k_selfatt(float* selfsc, float* maxE, const float* a_s, const float* a_d){
  int e = blockIdx.x * 256 + threadIdx.x;
  if (e < EE) { float v = lrelu(a_s[e] + a_d[e]); selfsc[e] = v; maxE[e] = v; }
}
__global__ void k_attmax(const int* src, const int* dst,
                         const float* a_s, const float* a_d, float* maxE){
  int l = blockIdx.x * 256 + threadIdx.x;
  if (l < LEE) atomicMaxF(&maxE[dst[l]], lrelu(a_s[src[l]] + a_d[dst[l]]));
}
__global__ void k_selfsum(float* sumE, const float* selfsc, const float* maxE){
  int e = blockIdx.x * 256 + threadIdx.x;
  if (e < EE) sumE[e] = __expf(selfsc[e] - maxE[e]);
}
__global__ void k_attsum(const int* src, const int* dst,
                         const float* a_s, const float* a_d,
                         const float* maxE, float* sumE){
  int l = blockIdx.x * 256 + threadIdx.x;
  if (l < LEE) {
    int d = dst[l];
    atomicAdd(&sumE[d], __expf(lrelu(a_s[src[l]] + a_d[d]) - maxE[d]));
  }
}
__global__ void k_wle(const int* src, const int* dst, const float* a_s, const float* a_d,
                      const float* maxE, const float* sumE, float* wl){
  int l = blockIdx.x * 256 + threadIdx.x;
  if (l < LEE) {
    int d = dst[l];
    wl[l] = __expf(lrelu(a_s[src[l]] + a_d[d]) - maxE[d]) / sumE[d];
  }
}
__global__ void k_hf_init(float* hf, const float* hw, const float* selfsc,
                          const float* maxE, const float* sumE, const float* gat_b){
  long long i = (long long)blockIdx.x * 256 + threadIdx.x;
  if (i < (long long)EE * FF) {
    int e = (int)(i / FF), f = (int)(i % FF);
    float w = __expf(selfsc[e] - maxE[e]) / sumE[e];
    hf[i] = w * hw[i] + gat_b[f];
  }
}
__global__ void k_hf_scatter(const int* src, const int* dst, const float* wl,
                             const float* hw, float* hf){
  long long i = (long long)blockIdx.x * 256 + threadIdx.x;
  if (i < (long long)LEE * FF) {
    int l = (int)(i / FF), f = (int)(i % FF);
    atomicAdd(&hf[(size_t)dst[l] * FF + f], wl[l] * hw[(size_t)src[l] * FF + f]);
  }
}
__global__ void k_gmax(const int* batch, const float* sc, float* gmax){
  int e = blockIdx.x * 256 + threadIdx.x;
  if (e < EE) atomicMaxF(&gmax[batch[e]], sc[e]);
}
__global__ void k_gsum(const int* batch, const float* sc, const float* gmax, float* gsum){
  int e = blockIdx.x * 256 + threadIdx.x;
  if (e < EE) { int b = batch[e]; atomicAdd(&gsum[b], __expf(sc[e] - gmax[b])); }
}
__global__ void k_scnorm(float* out, const int* batch, const float* sc,
                         const float* gmax, const float* gsum){
  int e = blockIdx.x * 256 + threadIdx.x;
  if (e < EE) { int b = batch[e]; out[e] = __expf(sc[e] - gmax[b]) / gsum[b]; }
}
__global__ void k_pool(const int* batch, const float* hf, const float* scE, float* pooled){
  long long i = (long long)blockIdx.x * 256 + threadIdx.x;
  if (i < (long long)EE * FF) {
    int e = (int)(i / FF), f = (int)(i % FF);
    atomicAdd(&pooled[(size_t)batch[e] * FF + f], hf[i] * scE[e]);
  }
}
__global__ void k_gscores(const float* gouts, const float* a, const float* a_bias, float* gsc){
  int g = blockIdx.x * 64 + threadIdx.x;
  if (g >= GGR) return;
  float s[NITER]; float mx = -3.4e38f;
  for (int t = 0; t < NITER; ++t) {
    float acc = a_bias[t];
    for (int f = 0; f < FF; ++f)
      acc += gouts[(size_t)t * GGR * FF + (size_t)g * FF + f] * a[f * NITER + t];
    s[t] = acc; mx = fmaxf(mx, acc);
  }
  float sum = 0.f;
  for (int t = 0; t < NITER; ++t) { s[t] = __expf(s[t] - mx); sum += s[t]; }
  for (int t = 0; t < NITER; ++t) gsc[g * NITER + t] = s[t] / sum;
}
__global__ void k_combine(float* comb, const float* outs, const float* gsc, const int* batch){
  long long i = (long long)blockIdx.x * 256 + threadIdx.x;
  if (i < (long long)EE * FF) {
    int e = (int)(i / FF);
    int b = batch[e];
    size_t EFt = (size_t)EE * FF;
    float acc = 0.f;
#pragma unroll
    for (int t = 0; t < NITER; ++t) acc += outs[t * EFt + i] * gsc[b * NITER + t];
    comb[i] = acc;
  }
}
__global__ void k_node_scatter(const int* ei1, const float* comb, float* xr){
  long long i = (long long)blockIdx.x * 256 + threadIdx.x;
  if (i < (long long)EE * FF) {
    int e = (int)(i / FF), f = (int)(i % FF);
    atomicAdd(&xr[(size_t)ei1[e] * FF + f], comb[i]);
  }
}
__global__ __launch_bounds__(256)
void k_bnstats(const float* X, float* mu, float* var, int M, int C){
  int c = blockIdx.x;
  float s = 0.f, s2 = 0.f;
  for (int r = threadIdx.x; r < M; r += 256) {
    float v = X[(size_t)r * C + c]; s += v; s2 += v * v;
  }
  __shared__ float sh[256], sh2[256];
  sh[threadIdx.x] = s; sh2[threadIdx.x] = s2; __syncthreads();
  for (int off = 128; off; off >>= 1) {
    if (threadIdx.x < off) { sh[threadIdx.x] += sh[threadIdx.x + off]; sh2[threadIdx.x] += sh2[threadIdx.x + off]; }
    __syncthreads();
  }
  if (threadIdx.x == 0) { float m = sh[0] / M; mu[c] = m; var[c] = sh2[0] / M - m * m; }
}
__global__ void k_bnapply(const float* X, const float* mu, const float* var,
                          const float* g, const float* b, const float* p,
                          float* Y, long long n, int C){
  long long i = (long long)blockIdx.x * 256 + threadIdx.x;
  if (i < n) {
    int c = (int)(i % C);
    float z = g[c] * (X[i] - mu[c]) * rsqrtf(var[c] + EPSV) + b[c];
    if (p) z = z >= 0.f ? z : p[0] * z;
    Y[i] = z;
  }
}

// ---------------------------------------------------------------------------
static inline unsigned nb(long long n){ return (unsigned)((n + 255) / 256); }

extern "C" void kernel_launch(void* const* d_in, const int* in_sizes, int n_in,
                              void* d_out, int out_size, void* d_ws, size_t ws_size,
                              hipStream_t stream)
{
  const float* x     = (const float*)d_in[0];
  const int*   ei    = (const int*)d_in[1];     // [2,E]
  const float* eattr = (const float*)d_in[2];
  const int*   lg    = (const int*)d_in[3];     // [2,LE]
  const int*   batch = (const int*)d_in[4];
  const float* W_u   = (const float*)d_in[5];
  const float* W_v   = (const float*)d_in[6];
  const float* W_e   = (const float*)d_in[7];
  const float* gcn_W = (const float*)d_in[8];
  const float* gcn_b = (const float*)d_in[9];
  const float* fbtl_W= (const float*)d_in[10];
  const float* fbtl_b= (const float*)d_in[11];
  const float* gat_W = (const float*)d_in[12];
  const float* gat_as= (const float*)d_in[13];
  const float* gat_ad= (const float*)d_in[14];
  const float* gat_b = (const float*)d_in[15];
  const float* a_p   = (const float*)d_in[16];
  const float* a_bias= (const float*)d_in[17];
  const float* gout_W= (const float*)d_in[18];
  const float* gout_b= (const float*)d_in[19];
  const float* bn1_g = (const float*)d_in[20];
  const float* bn1_b = (const float*)d_in[21];
  const float* W1    = (const float*)d_in[22];
  const float* b1    = (const float*)d_in[23];
  const float* bn2_g = (const float*)d_in[24];
  const float* bn2_b = (const float*)d_in[25];
  const float* p2    = (const float*)d_in[26];
  const float* W2    = (const float*)d_in[27];
  const float* b2    = (const float*)d_in[28];
  const float* bn3_g = (const float*)d_in[29];
  const float* bn3_b = (const float*)d_in[30];
  const float* p3    = (const float*)d_in[31];
  const float* W3    = (const float*)d_in[32];
  const float* b3    = (const float*)d_in[33];
  const float* bn4_g = (const float*)d_in[34];
  const float* bn4_b = (const float*)d_in[35];
  const float* p4    = (const float*)d_in[36];
  const float* W4    = (const float*)d_in[37];
  const float* b4    = (const float*)d_in[38];
  const float* bn5_g = (const float*)d_in[39];
  const float* bn5_b = (const float*)d_in[40];
  const float* p5    = (const float*)d_in[41];
  const float* W5    = (const float*)d_in[42];
  const float* b5    = (const float*)d_in[43];

  const int* ei0 = ei;            const int* ei1 = ei + EE;
  const int* lgs = lg;            const int* lgd = lg + LEE;

  // ---- workspace carve (floats) ----
  float* w = (float*)d_ws; size_t off = 0;
  auto alloc = [&](size_t n){ float* p = w + off; off += n; return p; };
  const size_t EF = (size_t)EE * FF;
  float* ea    = alloc(EF);
  float* msg   = alloc(EF);                 // also reused as combined out
  float* hw    = alloc(EF);
  float* hf    = alloc(EF);
  float* outs  = alloc((size_t)NITER * EF);
  float* xu    = alloc((size_t)NN * FF);    // also reused as bn(xr)
  float* xv    = alloc((size_t)NN * FF);
  float* xr    = alloc((size_t)NN * FF);
  float* B1    = alloc((size_t)NN * SNDD);
  float* B2    = alloc((size_t)NN * SNDD);
  float* B3    = alloc((size_t)NN * SNDD);
  float* B4    = alloc((size_t)NN * SNDD);
  float* dinv  = alloc(EE);
  float* o1    = alloc(EE);
  float* o2    = alloc(EE);
  float* scoreE= alloc(EE);
  float* a_s   = alloc(EE);
  float* a_d   = alloc(EE);
  float* selfsc= alloc(EE);
  float* maxE  = alloc(EE);
  float* sumE  = alloc(EE);
  float* scE   = alloc(EE);
  float* gcnacc= alloc(EE);
  float* wl    = alloc(LEE);
  float* gmax  = alloc(GGR);
  float* gsum  = alloc(GGR);
  float* pooled= alloc((size_t)GGR * FF);
  float* gouts = alloc((size_t)NITER * GGR * FF);
  float* gsc   = alloc((size_t)GGR * NITER);
  float* mu    = alloc(SNDD);
  float* var   = alloc(SNDD);
  (void)ws_size; (void)in_sizes; (void)n_in; (void)out_size;

  auto gemm = [&](const float* A, const float* Wm, const float* bias, float* C,
                  int M, int K, int N, int op){
    dim3 g((M + BM - 1) / BM, (N + BN - 1) / BN);
    k_gemm<<<g, 256, 0, stream>>>(A, Wm, bias, C, M, K, N, op);
  };

  // ---- edge features: ea = (xu[ei0] + xv[ei1] + eattr@W_e)/3 ----
  gemm(x, W_u, nullptr, xu, NN, FF, FF, 0);
  gemm(x, W_v, nullptr, xv, NN, FF, FF, 0);
  gemm(eattr, W_e, nullptr, ea, EE, EDIMM, FF, 0);
  k_build_ea<<<nb((long long)EE * FF), 256, 0, stream>>>(ea, xu, xv, ei0, ei1);

  // ---- GCN degree norm (structure-only, once) ----
  k_fill<<<nb(EE), 256, 0, stream>>>(dinv, 0.f, EE);
  k_deg<<<nb(LEE), 256, 0, stream>>>(lgd, dinv);
  k_dinv<<<nb(EE), 256, 0, stream>>>(dinv);

  // ---- message passing iterations ----
  for (int t = 0; t < NITER; ++t) {
    const float* prev = (t == 0) ? ea : outs + (size_t)(t - 1) * EF;
    float* outt = outs + (size_t)t * EF;
    k_fill<<<nb((long long)EF), 256, 0, stream>>>(msg, 0.f, (long long)EF);
    k_scatter_feat<<<nb((long long)LEE * FF), 256, 0, stream>>>(lgs, lgd, prev, msg);
    k_add<<<nb((long long)EF), 256, 0, stream>>>(outt, ea, msg, (long long)EF);

    // score = a*GCN + (1-a)*fbtl
    k_rowdot2<<<(EE + 7) / 8, 256, 0, stream>>>(outt, gcn_W, fbtl_W, o1, o2, EE, FF);
    k_fill<<<nb(EE), 256, 0, stream>>>(gcnacc, 0.f, EE);
    k_gcn_scatter<<<nb(LEE), 256, 0, stream>>>(lgs, lgd, dinv, o1, gcnacc);
    k_score<<<nb(EE), 256, 0, stream>>>(scoreE, gcnacc, dinv, o1, o2, gcn_b, fbtl_b);

    // GAT
    gemm(outt, gat_W, nullptr, hw, EE, FF, FF, 0);
    k_rowdot2<<<(EE + 7) / 8, 256, 0, stream>>>(hw, gat_as, gat_ad, a_s, a_d, EE, FF);
    k_selfatt<<<nb(EE), 256, 0, stream>>>(selfsc, maxE, a_s, a_d);
    k_attmax<<<nb(LEE), 256, 0, stream>>>(lgs, lgd, a_s, a_d, maxE);
    k_selfsum<<<nb(EE), 256, 0, stream>>>(sumE, selfsc, maxE);
    k_attsum<<<nb(LEE), 256, 0, stream>>>(lgs, lgd, a_s, a_d, maxE, sumE);
    k_wle<<<nb(LEE), 256, 0, stream>>>(lgs, lgd, a_s, a_d, maxE, sumE, wl);
    k_hf_init<<<nb((long long)EF), 256, 0, stream>>>(hf, hw, selfsc, maxE, sumE, gat_b);
    k_hf_scatter<<<nb((long long)LEE * FF), 256, 0, stream>>>(lgs, lgd, wl, hw, hf);

    // per-graph softmax of score, pool
    k_fill<<<1, 256, 0, stream>>>(gmax, -3.4e38f, GGR);
    k_gmax<<<nb(EE), 256, 0, stream>>>(batch, scoreE, gmax);
    k_fill<<<1, 256, 0, stream>>>(gsum, 0.f, GGR);
    k_gsum<<<nb(EE), 256, 0, stream>>>(batch, scoreE, gmax, gsum);
    k_scnorm<<<nb(EE), 256, 0, stream>>>(scE, batch, scoreE, gmax, gsum);
    k_fill<<<nb((long long)GGR * FF), 256, 0, stream>>>(pooled, 0.f, (long long)GGR * FF);
    k_pool<<<nb((long long)EF), 256, 0, stream>>>(batch, hf, scE, pooled);
    gemm(pooled, gout_W, gout_b, gouts + (size_t)t * GGR * FF, GGR, FF, FF, 1 /*tanh*/);
  }

  // ---- iteration-attention combine, node scatter ----
  k_gscores<<<(GGR + 63) / 64, 64, 0, stream>>>(gouts, a_p, a_bias, gsc);
  k_combine<<<nb((long long)EF), 256, 0, stream>>>(msg, outs, gsc, batch);
  k_copy<<<nb((long long)NN * FF), 256, 0, stream>>>(xr, x, (long long)NN * FF);
  k_node_scatter<<<nb((long long)EF), 256, 0, stream>>>(ei1, msg, xr);

  // ---- MLP tail ----
  k_bnstats<<<FF, 256, 0, stream>>>(xr, mu, var, NN, FF);
  k_bnapply<<<nb((long long)NN * FF), 256, 0, stream>>>(xr, mu, var, bn1_g, bn1_b, nullptr, xu, (long long)NN * FF, FF);
  gemm(xu, W1, b1, B1, NN, FF, SNDD, 0);                       // h1

  k_bnstats<<<SNDD, 256, 0, stream>>>(B1, mu, var, NN, SNDD);
  k_bnapply<<<nb((long long)NN * SNDD), 256, 0, stream>>>(B1, mu, var, bn2_g, bn2_b, p2, B2, (long long)NN * SNDD, SNDD);
  gemm(B2, W2, b2, B3, NN, SNDD, SNDD, 0);                     // h2

  k_bnstats<<<SNDD, 256, 0, stream>>>(B3, mu, var, NN, SNDD);
  k_bnapply<<<nb((long long)NN * SNDD), 256, 0, stream>>>(B3, mu, var, bn3_g, bn3_b, p3, B2, (long long)NN * SNDD, SNDD);
  gemm(B2, W3, b3, B4, NN, SNDD, SNDD, 0);                     // h3
  k_avg2<<<nb((long long)NN * SNDD), 256, 0, stream>>>(B1, B4, (long long)NN * SNDD);  // h = (h3+h1)/2

  k_bnstats<<<SNDD, 256, 0, stream>>>(B1, mu, var, NN, SNDD);
  k_bnapply<<<nb((long long)NN * SNDD), 256, 0, stream>>>(B1, mu, var, bn4_g, bn4_b, p4, B2, (long long)NN * SNDD, SNDD);
  gemm(B2, W4, b4, B3, NN, SNDD, SNDD, 0);                     // h4
  k_avg2<<<nb((long long)NN * SNDD), 256, 0, stream>>>(B1, B3, (long long)NN * SNDD);  // h = (h4+h)/2

  k_bnstats<<<SNDD, 256, 0, stream>>>(B1, mu, var, NN, SNDD);
  k_bnapply<<<nb((long long)NN * SNDD), 256, 0, stream>>>(B1, mu, var, bn5_g, bn5_b, p5, B2, (long long)NN * SNDD, SNDD);
  gemm(B2, W5, b5, (float*)d_out, NN, SNDD, FF, 0);            // final [N,F]
}